// Inception_LocalNonLocal_v2_79293686219341
// MI455X (gfx1250) — compile-verified
//
#include <hip/hip_runtime.h>
#include <hip/hip_bf16.h>
#include <math.h>

// ---------------------------------------------------------------------------
// Problem constants (fixed by the reference)
// ---------------------------------------------------------------------------
#define BATCH 4
#define CIN   256
#define CPD   32              // attention channels (C')
#define HH    64
#define WW    64
#define NPIX  (HH * WW)       // 4096 pixels per image
#define KQX_C 96              // stacked k(0..31) / q(32..63) / x(64..95)
#define BN_EPS 1e-5f

typedef __attribute__((ext_vector_type(2))) float v2f;
typedef __attribute__((ext_vector_type(8))) float v8f;

// ---------------------------------------------------------------------------
// Kernel A: fused k/q/x 1x1-conv projection.
//   D[96 x 4096] = W[96 x 256] * X[256 x 4096] per batch via
//   V_WMMA_F32_16X16X4_F32. Each wave owns ONE 16-pixel column tile and all
//   SIX 16-row M-tiles: one B-fragment load feeds 6 independent WMMAs
//   (6x reuse of the large X operand, no inter-WMMA RAW hazards).
//   Epilogue: +bias, vessel mask on key rows, store pixel-major [b][pix][96].
// ---------------------------------------------------------------------------
__global__ __launch_bounds__(128) void proj_wmma_kernel(
    const float* __restrict__ x, const float* __restrict__ vessel,
    const float* __restrict__ wk, const float* __restrict__ bk,
    const float* __restrict__ wq, const float* __restrict__ bq,
    const float* __restrict__ wx, const float* __restrict__ bx,
    float* __restrict__ kqx)
{
  const int wave = (blockIdx.x * blockDim.x + threadIdx.x) >> 5;
  const int lane = threadIdx.x & 31;

  const int b  = wave / (NPIX / 16);            // batch
  const int p0 = (wave % (NPIX / 16)) * 16;     // pixel tile base

  const int l15 = lane & 15;
  const int hi  = lane >> 4;
  const int kb  = hi * 2;                       // A/B fragment K sub-offset

  // A-fragment row pointers for the 6 M-tiles of the stacked [96,256] weight:
  //   j=0,1 -> wk rows, j=2,3 -> wq rows, j=4,5 -> wx rows
  const float* wrow[6];
  wrow[0] = wk + (l15)      * CIN;
  wrow[1] = wk + (16 + l15) * CIN;
  wrow[2] = wq + (l15)      * CIN;
  wrow[3] = wq + (16 + l15) * CIN;
  wrow[4] = wx + (l15)      * CIN;
  wrow[5] = wx + (16 + l15) * CIN;

  // B-fragment source: column (pixel) of x, channel-major [B][C][H*W]
  const float* xcol = x + (size_t)b * CIN * NPIX + (p0 + l15);

  v8f acc[6];
  #pragma unroll
  for (int j = 0; j < 6; ++j) acc[j] = (v8f){0.f,0.f,0.f,0.f,0.f,0.f,0.f,0.f};

  #pragma unroll 2
  for (int k0 = 0; k0 < CIN; k0 += 4) {
    v2f bm;
    bm.x = xcol[(size_t)(k0 + kb) * NPIX];      // B[k0+kb][n]
    bm.y = xcol[(size_t)(k0 + kb + 1) * NPIX];  // B[k0+kb+1][n]
    #pragma unroll
    for (int j = 0; j < 6; ++j) {
      v2f a = *(const v2f*)(wrow[j] + k0 + kb); // A[m][k0+kb..+1]
      acc[j] = __builtin_amdgcn_wmma_f32_16x16x4_f32(
          /*neg_a=*/false, a, /*neg_b=*/false, bm,
          /*c_mod=*/(short)0, acc[j], /*reuse_a=*/false, /*reuse_b=*/false);
    }
  }

  const float vsl = vessel[(size_t)b * NPIX + p0 + l15];  // column is fixed per lane
  float* outpix = kqx + ((size_t)b * NPIX + p0 + l15) * KQX_C;

  #pragma unroll
  for (int j = 0; j < 6; ++j) {
    const float* barr = (j < 2) ? bk : (j < 4 ? bq : bx);
    const int    boff = (j & 1) * 16;
    #pragma unroll
    for (int v = 0; v < 8; ++v) {
      const int sub = v + 8 * hi;               // D row-in-tile per ISA layout
      float val = acc[j][v] + barr[boff + sub];
      if (j < 2) val *= vsl;                    // key map masked by vessel
      outpix[j * 16 + sub] = val;
    }
  }
}

// ---------------------------------------------------------------------------
// Kernel B: multi-scale local attention, one thread per pixel.
//   Per scale: two-pass softmax over the k*k window. Zero-padded OOB taps
//   contribute logit 0 to max/Z but no value (matches dilated-patches padding).
//   Pass 1 prefetches the xm lines (256B into the pixel record) for pass 2.
// ---------------------------------------------------------------------------
__global__ __launch_bounds__(256) void attn_kernel(
    const float* __restrict__ kqx, float* __restrict__ pre)
{
  const int g  = blockIdx.x * blockDim.x + threadIdx.x;   // [0, B*NPIX)
  const int b  = g / NPIX;
  const int p  = g % NPIX;
  const int py = p / WW, px = p % WW;
  const float* base = kqx + (size_t)b * NPIX * KQX_C;

  float4 q[8];
  {
    const float4* qp = (const float4*)(base + (size_t)p * KQX_C + 32);
    #pragma unroll
    for (int i = 0; i < 8; ++i) q[i] = qp[i];
  }

  float4 o[8];
  #pragma unroll
  for (int i = 0; i < 8; ++i) o[i] = make_float4(0.f, 0.f, 0.f, 0.f);

  for (int r = 1; r <= 4; ++r) {                // scales k = 3,5,7,9
    // ---- pass 1: window max (OOB logit is exactly 0) ----
    float mmax = -3.0e38f;
    for (int dy = -r; dy <= r; ++dy) {
      for (int dx = -r; dx <= r; ++dx) {
        const int yy = py + dy, xx = px + dx;
        float s = 0.f;
        if (yy >= 0 && yy < HH && xx >= 0 && xx < WW) {
          const float* pix = base + (size_t)(yy * WW + xx) * KQX_C;
          __builtin_prefetch(pix + 64, 0, 3);   // pull xm lines for pass 2
          const float4* kp = (const float4*)pix;
          #pragma unroll
          for (int i = 0; i < 8; ++i) {
            float4 kv = kp[i];
            s += kv.x * q[i].x + kv.y * q[i].y + kv.z * q[i].z + kv.w * q[i].w;
          }
        }
        mmax = fmaxf(mmax, s);
      }
    }
    // ---- pass 2: exp-sum + value accumulation ----
    float Z = 0.f;
    const float eoob = __expf(-mmax);
    float4 a[8];
    #pragma unroll
    for (int i = 0; i < 8; ++i) a[i] = make_float4(0.f, 0.f, 0.f, 0.f);

    for (int dy = -r; dy <= r; ++dy) {
      for (int dx = -r; dx <= r; ++dx) {
        const int yy = py + dy, xx = px + dx;
        if (yy >= 0 && yy < HH && xx >= 0 && xx < WW) {
          const float* pix = base + (size_t)(yy * WW + xx) * KQX_C;
          const float4* kp = (const float4*)pix;
          float s = 0.f;
          #pragma unroll
          for (int i = 0; i < 8; ++i) {
            float4 kv = kp[i];
            s += kv.x * q[i].x + kv.y * q[i].y + kv.z * q[i].z + kv.w * q[i].w;
          }
          const float e = __expf(s - mmax);
          Z += e;
          const float4* xp = (const float4*)(pix + 64);
          #pragma unroll
          for (int i = 0; i < 8; ++i) {
            float4 xv = xp[i];
            a[i].x += e * xv.x; a[i].y += e * xv.y;
            a[i].z += e * xv.z; a[i].w += e * xv.w;
          }
        } else {
          Z += eoob;                             // padded tap: logit 0, value 0
        }
      }
    }
    const float inv = 1.f / Z;
    #pragma unroll
    for (int i = 0; i < 8; ++i) {
      o[i].x += a[i].x * inv; o[i].y += a[i].y * inv;
      o[i].z += a[i].z * inv; o[i].w += a[i].w * inv;
    }
  }

  float4* op = (float4*)(pre + (size_t)g * CPD);
  #pragma unroll
  for (int i = 0; i < 8; ++i) op[i] = o[i];
}

// ---------------------------------------------------------------------------
// Kernel C: final 1x1 conv y = wf[256,32] * pre + bf (WMMA f32).
//   Each wave owns one 16-pixel column tile and FOUR M-tiles (64 rows):
//   one B-fragment load feeds 4 independent WMMAs. Per-channel sum / sum^2
//   for BatchNorm via shfl_xor butterfly + 2 f32 atomics per D row.
// ---------------------------------------------------------------------------
__global__ __launch_bounds__(128) void outproj_wmma_kernel(
    const float* __restrict__ pre, const float* __restrict__ wf,
    const float* __restrict__ bf, float* __restrict__ y,
    float* __restrict__ stats)
{
  const int wave = (blockIdx.x * blockDim.x + threadIdx.x) >> 5;
  const int lane = threadIdx.x & 31;

  const int tiles_per_b = 4 * (NPIX / 16);      // 4 M-groups x 256 N-tiles
  const int b   = wave / tiles_per_b;
  const int rem = wave % tiles_per_b;
  const int mg  = rem / (NPIX / 16);            // M-group: rows [mg*64, mg*64+64)
  const int p0  = (rem % (NPIX / 16)) * 16;

  const int l15 = lane & 15;
  const int hi  = lane >> 4;
  const int kb  = hi * 2;

  const float* arow0 = wf + (mg * 64 + l15) * CPD;                   // [256][32]
  const float* brow  = pre + ((size_t)b * NPIX + p0 + l15) * CPD;    // pixel-major

  v8f acc[4];
  #pragma unroll
  for (int j = 0; j < 4; ++j) acc[j] = (v8f){0.f,0.f,0.f,0.f,0.f,0.f,0.f,0.f};

  #pragma unroll
  for (int k0 = 0; k0 < CPD; k0 += 4) {
    v2f bm = *(const v2f*)(brow + k0 + kb);
    #pragma unroll
    for (int j = 0; j < 4; ++j) {
      v2f a = *(const v2f*)(arow0 + j * 16 * CPD + k0 + kb);
      acc[j] = __builtin_amdgcn_wmma_f32_16x16x4_f32(
          false, a, false, bm, (short)0, acc[j], false, false);
    }
  }

  const int col = p0 + l15;
  #pragma unroll
  for (int j = 0; j < 4; ++j) {
    #pragma unroll
    for (int v = 0; v < 8; ++v) {
      const int row = mg * 64 + j * 16 + v + 8 * hi;
      const float val = acc[j][v] + bf[row];
      y[((size_t)b * CIN + row) * NPIX + col] = val;                 // coalesced

      // reduce across the 16 columns held by this half-wave
      float s = val, ss = val * val;
      #pragma unroll
      for (int m = 1; m <= 8; m <<= 1) {
        s  += __shfl_xor(s,  m, 32);
        ss += __shfl_xor(ss, m, 32);
      }
      if (l15 == 0) {
        atomicAdd(stats + row, s);
        atomicAdd(stats + CIN + row, ss);
      }
    }
  }
}

// ---------------------------------------------------------------------------
// Kernel D: BatchNorm (batch stats) + residual, float4 vectorized.
// ---------------------------------------------------------------------------
__global__ __launch_bounds__(256) void bn_residual_kernel(
    const float* __restrict__ x, const float* __restrict__ y,
    const float* __restrict__ stats, const float* __restrict__ gamma,
    const float* __restrict__ beta, float* __restrict__ out)
{
  const int i4 = blockIdx.x * blockDim.x + threadIdx.x;   // float4 index
  const int total4 = BATCH * CIN * NPIX / 4;
  if (i4 >= total4) return;

  const int e = i4 * 4;
  const int c = (e / NPIX) % CIN;                          // NPIX % 4 == 0

  const float cnt  = (float)(BATCH * NPIX);
  const float mean = stats[c] / cnt;
  const float var  = stats[CIN + c] / cnt - mean * mean;   // biased (ddof=0)
  const float gsc  = gamma[c] * rsqrtf(var + BN_EPS);
  const float bt   = beta[c];

  const float4 xv = ((const float4*)x)[i4];
  const float4 yv = ((const float4*)y)[i4];
  float4 ov;
  ov.x = xv.x + (yv.x - mean) * gsc + bt;
  ov.y = xv.y + (yv.y - mean) * gsc + bt;
  ov.z = xv.z + (yv.z - mean) * gsc + bt;
  ov.w = xv.w + (yv.w - mean) * gsc + bt;
  ((float4*)out)[i4] = ov;
}

// ---------------------------------------------------------------------------
// Host-side launcher (graph-capture safe: only stream ops).
// ---------------------------------------------------------------------------
extern "C" void kernel_launch(void* const* d_in, const int* in_sizes, int n_in,
                              void* d_out, int out_size, void* d_ws, size_t ws_size,
                              hipStream_t stream) {
  const float* x      = (const float*)d_in[0];
  const float* vessel = (const float*)d_in[1];
  const float* wk     = (const float*)d_in[2];
  const float* bk     = (const float*)d_in[3];
  const float* wq     = (const float*)d_in[4];
  const float* bq     = (const float*)d_in[5];
  const float* wx     = (const float*)d_in[6];
  const float* bx     = (const float*)d_in[7];
  const float* wf     = (const float*)d_in[8];
  const float* bf     = (const float*)d_in[9];
  const float* gamma  = (const float*)d_in[10];
  const float* beta   = (const float*)d_in[11];
  float* out = (float*)d_out;

  // Workspace layout (floats): kqx | pre | y | stats   (~25.2 MB total)
  float* kqx   = (float*)d_ws;                               // B*4096*96
  float* pre   = kqx + (size_t)BATCH * NPIX * KQX_C;         // B*4096*32
  float* y     = pre + (size_t)BATCH * NPIX * CPD;           // B*256*4096
  float* stats = y   + (size_t)BATCH * CIN * NPIX;           // 512

  hipMemsetAsync(stats, 0, 2 * CIN * sizeof(float), stream);

  // A: 1024 wave-tiles (one per 16-pixel column, 6 M-tiles each), 4 waves/block
  proj_wmma_kernel<<<BATCH * (NPIX / 16) / 4, 128, 0, stream>>>(
      x, vessel, wk, bk, wq, bq, wx, bx, kqx);

  // B: one thread per pixel
  attn_kernel<<<(BATCH * NPIX) / 256, 256, 0, stream>>>(kqx, pre);

  // C: 4096 wave-tiles (one per 16-pixel column per 64-row M-group)
  outproj_wmma_kernel<<<BATCH * 4 * (NPIX / 16) / 4, 128, 0, stream>>>(
      pre, wf, bf, y, stats);

  // D: float4 elementwise
  bn_residual_kernel<<<(BATCH * CIN * NPIX / 4) / 256, 256, 0, stream>>>(
      x, y, stats, gamma, beta, out);
}